// DGM_module_58308476011165
// MI455X (gfx1250) — compile-verified
//
#include <hip/hip_runtime.h>
#include <math.h>

// Problem constants (match reference)
#define BATCH   16
#define NNODES  2048
#define DIN     64
#define DH      128
#define KNN     8
#define NTOT    (BATCH * NNODES)   // 32768

typedef __attribute__((ext_vector_type(2))) float v2f;
typedef __attribute__((ext_vector_type(4))) float v4f;
typedef __attribute__((ext_vector_type(8))) float v8f;

__device__ __forceinline__ v2f ld2(const float* p) { return *(const v2f*)p; }
__device__ __forceinline__ v4f ld4(const float* p) { return *(const v4f*)p; }

__device__ __forceinline__ float temp_scale(const float* tptr) {
  float tv = *tptr;
  tv = fminf(fmaxf(tv, -5.0f), 5.0f);
  return expf(tv);
}

// ---------------------------------------------------------------------------
// fp32 WMMA GEMM: out[M x 128] = act(A[M x Kd] @ W[Kd x 128] + bias)
// One wave per 16x64 output strip (4 N-subtiles), V_WMMA_F32_16X16X4_F32.
// A float2 load amortized over 4 WMMAs. Grid supplies M/16 * 2 waves.
// ---------------------------------------------------------------------------
__global__ void gemm_bias_act(const float* __restrict__ A,
                              const float* __restrict__ W,
                              const float* __restrict__ bias,
                              float* __restrict__ out,
                              int Kd, int leaky) {
  const int wave = blockIdx.x * (blockDim.x >> 5) + (threadIdx.x >> 5);
  const int I    = wave >> 1;           // row tile
  const int JB   = wave & 1;            // 64-col block (128/64 = 2)
  const int lane = threadIdx.x & 31;
  const int half = lane >> 4;
  const int l    = lane & 15;

  const float* arow = A + (size_t)(I * 16 + l) * Kd;
  const float* wcol[4];
#pragma unroll
  for (int jj = 0; jj < 4; ++jj)
    wcol[jj] = W + (JB * 64 + jj * 16 + l);          // B[k][n] column base

  v8f acc[4] = {};
  for (int k0 = 0; k0 < Kd; k0 += 4) {
    const int kk = k0 + 2 * half;
    v2f a = ld2(arow + kk);                          // A[l][kk], A[l][kk+1]
#pragma unroll
    for (int jj = 0; jj < 4; ++jj) {
      v2f b;
      b.x = wcol[jj][(size_t)(kk + 0) * DH];
      b.y = wcol[jj][(size_t)(kk + 1) * DH];
      acc[jj] = __builtin_amdgcn_wmma_f32_16x16x4_f32(false, a, false, b,
                                                      (short)0, acc[jj],
                                                      false, false);
    }
  }

#pragma unroll
  for (int jj = 0; jj < 4; ++jj) {
    const int col = JB * 64 + jj * 16 + l;
    const float bs = bias[col];
#pragma unroll
    for (int v = 0; v < 8; ++v) {
      const int row = I * 16 + v + (half << 3);
      float val = acc[jj][v] + bs;
      if (leaky) val = val > 0.0f ? val : 0.1f * val;
      out[(size_t)row * DH + col] = val;
    }
  }
}

// ---------------------------------------------------------------------------
// Row squared-norms: out[r] = sum_c A[r][c]^2   (cols multiple of 4)
// ---------------------------------------------------------------------------
__global__ void rownorm(const float* __restrict__ A, float* __restrict__ out,
                        int rows, int cols) {
  const int r = blockIdx.x * blockDim.x + threadIdx.x;
  if (r >= rows) return;
  const float* p = A + (size_t)r * cols;
  float s = 0.0f;
  for (int c = 0; c < cols; c += 4) {
    v4f v = ld4(p + c);
    s += v.x * v.x + v.y * v.y + v.z * v.z + v.w * v.w;
  }
  out[r] = s;
}

// ---------------------------------------------------------------------------
// lq[i][j] = t * sum_b max(sq[b][i] + sq[b][j] - 2 * <emb_b[i], emb_b[j]>, 0)
// One wave per 16x64 strip: 16 batches x (32 K-steps x 4 WMMA).
// ---------------------------------------------------------------------------
__global__ void lq_kernel(const float* __restrict__ emb,
                          const float* __restrict__ sq,
                          const float* __restrict__ tptr,
                          float* __restrict__ lq) {
  const int wave = blockIdx.x * (blockDim.x >> 5) + (threadIdx.x >> 5);
  const int I    = wave >> 5;           // 128 row tiles
  const int JB   = wave & 31;           // 32 blocks of 64 cols
  const int lane = threadIdx.x & 31;
  const int half = lane >> 4;
  const int l    = lane & 15;
  const float t  = temp_scale(tptr);

  v8f acc[4] = {};
  for (int b = 0; b < BATCH; ++b) {
    const float* E  = emb + (size_t)b * NNODES * DH;
    const float* ra = E + (size_t)(I * 16 + l) * DH;
    const float* rb[4];
#pragma unroll
    for (int jj = 0; jj < 4; ++jj)
      rb[jj] = E + (size_t)(JB * 64 + jj * 16 + l) * DH;

    v8f g[4] = {};
#pragma unroll 4
    for (int k0 = 0; k0 < DH; k0 += 4) {
      const int kk = k0 + 2 * half;
      v2f a = ld2(ra + kk);
#pragma unroll
      for (int jj = 0; jj < 4; ++jj) {
        v2f bb = ld2(rb[jj] + kk);
        g[jj] = __builtin_amdgcn_wmma_f32_16x16x4_f32(false, a, false, bb,
                                                      (short)0, g[jj],
                                                      false, false);
      }
    }
    const float* sqb = sq + b * NNODES;
#pragma unroll
    for (int jj = 0; jj < 4; ++jj) {
      const float sj = sqb[JB * 64 + jj * 16 + l];
#pragma unroll
      for (int v = 0; v < 8; ++v) {
        const int mi = I * 16 + v + (half << 3);
        float d = sqb[mi] + sj - 2.0f * g[jj][v];
        acc[jj][v] += fmaxf(d, 0.0f);
      }
    }
  }
#pragma unroll
  for (int jj = 0; jj < 4; ++jj) {
    const int col = JB * 64 + jj * 16 + l;
#pragma unroll
    for (int v = 0; v < 8; ++v) {
      const int mi = I * 16 + v + (half << 3);
      lq[(size_t)mi * NNODES + col] = acc[jj][v] * t;
    }
  }
}

// ---------------------------------------------------------------------------
// D2[i][j] = max(nlq[i] + nlq[j] - 2 * <lq[i], lq[j]>, 0)   (K = 2048)
// One wave per 16x64 strip; 512 K-steps x 4 WMMA, A-load amortized 4x.
// ---------------------------------------------------------------------------
__global__ void d2_kernel(const float* __restrict__ lq,
                          const float* __restrict__ nlq,
                          float* __restrict__ D2) {
  const int wave = blockIdx.x * (blockDim.x >> 5) + (threadIdx.x >> 5);
  const int I    = wave >> 5;
  const int JB   = wave & 31;
  const int lane = threadIdx.x & 31;
  const int half = lane >> 4;
  const int l    = lane & 15;

  const float* ra = lq + (size_t)(I * 16 + l) * NNODES;
  const float* rb[4];
#pragma unroll
  for (int jj = 0; jj < 4; ++jj)
    rb[jj] = lq + (size_t)(JB * 64 + jj * 16 + l) * NNODES;

  v8f g[4] = {};
#pragma unroll 4
  for (int k0 = 0; k0 < NNODES; k0 += 4) {
    const int kk = k0 + 2 * half;
    v2f a = ld2(ra + kk);
#pragma unroll
    for (int jj = 0; jj < 4; ++jj) {
      v2f bb = ld2(rb[jj] + kk);
      g[jj] = __builtin_amdgcn_wmma_f32_16x16x4_f32(false, a, false, bb,
                                                    (short)0, g[jj],
                                                    false, false);
    }
  }

#pragma unroll
  for (int jj = 0; jj < 4; ++jj) {
    const int col = JB * 64 + jj * 16 + l;
    const float nj = nlq[col];
#pragma unroll
    for (int v = 0; v < 8; ++v) {
      const int mi = I * 16 + v + (half << 3);
      D2[(size_t)mi * NNODES + col] =
          fmaxf(nlq[mi] + nj - 2.0f * g[jj][v], 0.0f);
    }
  }
}

// ---------------------------------------------------------------------------
// Stable top-K smallest per row (ties -> lowest index first, like jax top_k).
// ---------------------------------------------------------------------------
__global__ void topk_kernel(const float* __restrict__ D2, int* __restrict__ idx) {
  const int r = blockIdx.x * blockDim.x + threadIdx.x;
  if (r >= NNODES) return;
  float bd[KNN];
  int   bj[KNN];
#pragma unroll
  for (int k = 0; k < KNN; ++k) { bd[k] = 3.4e38f; bj[k] = 0; }
  const float* row = D2 + (size_t)r * NNODES;
  for (int j = 0; j < NNODES; ++j) {
    const float d = row[j];
    if (d < bd[KNN - 1]) {           // strict < keeps earlier index on ties
      int p = KNN - 1;
      while (p > 0 && d < bd[p - 1]) {
        bd[p] = bd[p - 1]; bj[p] = bj[p - 1]; --p;
      }
      bd[p] = d; bj[p] = j;
    }
  }
#pragma unroll
  for (int k = 0; k < KNN; ++k) idx[r * KNN + k] = bj[k];
}

// ---------------------------------------------------------------------------
// edges (b-major [2, B*N*K], indices as float) + logprobs[b][n][k]
// logprob = -t * || emb[idx[n][k]] - emb[b*N + n] ||^2   (gather from batch 0)
// ---------------------------------------------------------------------------
__global__ void edge_lp_kernel(const float* __restrict__ emb,
                               const int* __restrict__ idx,
                               const float* __restrict__ tptr,
                               float* __restrict__ edges,
                               float* __restrict__ lp) {
  const int L   = BATCH * NNODES * KNN;
  const int gid = blockIdx.x * blockDim.x + threadIdx.x;
  if (gid >= L) return;
  const int b   = gid / (NNODES * KNN);
  const int rem = gid - b * (NNODES * KNN);
  const int n   = rem / KNN;
  const int k   = rem - n * KNN;
  const int tj  = idx[n * KNN + k];

  edges[gid]     = (float)(n + b * NNODES);
  edges[L + gid] = (float)(tj + b * NNODES);

  const float* p = emb + (size_t)tj * DH;                  // batch-0 gather
  const float* q = emb + (size_t)(b * NNODES + n) * DH;
  float s = 0.0f;
  for (int c = 0; c < DH; c += 4) {
    v4f a = ld4(p + c), bb = ld4(q + c);
    float dx = a.x - bb.x, dy = a.y - bb.y, dz = a.z - bb.z, dw = a.w - bb.w;
    s += dx * dx + dy * dy + dz * dz + dw * dw;
  }
  lp[gid] = -s * temp_scale(tptr);
}

// ---------------------------------------------------------------------------
extern "C" void kernel_launch(void* const* d_in, const int* in_sizes, int n_in,
                              void* d_out, int out_size, void* d_ws, size_t ws_size,
                              hipStream_t stream) {
  (void)in_sizes; (void)n_in; (void)out_size; (void)ws_size;
  const float* x  = (const float*)d_in[0];
  const float* W1 = (const float*)d_in[1];
  const float* b1 = (const float*)d_in[2];
  const float* W2 = (const float*)d_in[3];
  const float* b2 = (const float*)d_in[4];
  const float* tp = (const float*)d_in[5];

  float* out   = (float*)d_out;
  float* xe    = out;                                   // [32768,128] == emb
  float* edges = out + (size_t)NTOT * DH;               // [2, B*N*K] as float
  float* lp    = edges + (size_t)2 * BATCH * NNODES * KNN;

  char*  w    = (char*)d_ws;
  float* hbuf = (float*)w;                                          // 16 MB
  float* lq   = (float*)(w + (size_t)NTOT * DH * sizeof(float));    // 16 MB
  float* sq   = lq + (size_t)NNODES * NNODES;                       // 32768
  float* nlq  = sq + NTOT;                                          // 2048
  int*   idx  = (int*)(nlq + NNODES);                               // 2048*8
  float* D2   = hbuf;   // reuse h buffer once gemm2 is done

  // 1) h = leaky_relu(x @ W1 + b1) : (32768/16)*2 = 4096 waves
  gemm_bias_act<<<1024, 128, 0, stream>>>(x, W1, b1, hbuf, DIN, 1);
  // 2) emb = h @ W2 + b2 -> xe (emb lives in d_out)
  gemm_bias_act<<<1024, 128, 0, stream>>>(hbuf, W2, b2, xe, DH, 0);
  // 3) per-row squared norms of emb
  rownorm<<<NTOT / 256, 256, 0, stream>>>(xe, sq, NTOT, DH);
  // 4) lq = t * sum_b clamp(pairwise_sq(emb_b)) : 128*32 = 4096 waves
  lq_kernel<<<1024, 128, 0, stream>>>(xe, sq, tp, lq);
  // 5) row norms of lq
  rownorm<<<NNODES / 256, 256, 0, stream>>>(lq, nlq, NNODES, NNODES);
  // 6) D2 = pairwise_sq(lq) : 128*32 = 4096 waves
  d2_kernel<<<1024, 128, 0, stream>>>(lq, nlq, D2);
  // 7) stable 8-NN per row
  topk_kernel<<<NNODES / 256, 256, 0, stream>>>(D2, idx);
  // 8) edges + logprobs
  edge_lp_kernel<<<(BATCH * NNODES * KNN) / 256, 256, 0, stream>>>(
      xe, idx, tp, edges, lp);
}